// GeodesicVectorQuantizer_56899726737761
// MI455X (gfx1250) — compile-verified
//
#include <hip/hip_runtime.h>

// VQ quantizer for MI455X (gfx1250, wave32).
// Outputs (concatenated in d_out, fp32 view):
//   [0 .. 16777215]            quantized_st (== codebook[argmin]) 65536x256
//   [16777216]                 quantization_loss = 1.25 * mean((q-x)^2)
//   [16777217 .. 16777217+65535] encoding_indices (stored as float values)

typedef __attribute__((ext_vector_type(2))) float v2f;
typedef __attribute__((ext_vector_type(8))) float v8f;

#define NROWS   65536
#define DIM     256
#define KCODES  1024

// ---------------- kernel 1: c2[n] = ||codebook[n]||^2 ----------------
__global__ __launch_bounds__(256) void vq_c2_kernel(const float* __restrict__ cb,
                                                    float* __restrict__ c2) {
    int n = blockIdx.x * 256 + threadIdx.x;           // 4 blocks x 256 = 1024
    if (n >= KCODES) return;
    const float4* row = (const float4*)(cb + (size_t)n * DIM);
    float s = 0.0f;
#pragma unroll
    for (int i = 0; i < DIM / 4; ++i) {
        float4 v = row[i];
        s += v.x * v.x + v.y * v.y + v.z * v.z + v.w * v.w;
    }
    c2[n] = s;
}

// ---------------- kernel 2: WMMA f32 distances + running argmin ----------------
// Block = 256 threads = 8 waves. Each wave owns a 16-row M-tile of x.
// Per wave: A fragments (16x256 x-tile) resident in VGPRs; loop 32 iterations,
// each computing two 16x16 dot tiles (32 codes) via chained V_WMMA_F32_16X16X4_F32.
__global__ __launch_bounds__(256) void vq_argmin_kernel(const float* __restrict__ x,
                                                        const float* __restrict__ cb,
                                                        const float* __restrict__ c2,
                                                        float* __restrict__ out_idx_f) {
    const int lane = threadIdx.x & 31;
    const int wave = threadIdx.x >> 5;
    const int lr   = lane & 15;          // row-within-tile (A) / column code (B)
    const int h    = lane >> 4;          // K-half selector: K={0,1} vs {2,3}
    const int rowBase = blockIdx.x * 128 + wave * 16;

    // --- load A fragments: a[kk] = { x[row][4kk+2h], x[row][4kk+2h+1] } ---
    const float* xrow = x + (size_t)(rowBase + lr) * DIM + 2 * h;
    v2f a[64];
#pragma unroll
    for (int kk = 0; kk < 64; ++kk)
        a[kk] = *(const v2f*)(xrow + 4 * kk);

    float minv[8];
    int   mini[8];
#pragma unroll
    for (int r = 0; r < 8; ++r) { minv[r] = 3.4e38f; mini[r] = 0; }

    for (int nt = 0; nt < 32; ++nt) {
        const int n0 = nt * 32 + lr;     // this lane's code column, tile 0
        const int n1 = n0 + 16;          // tile 1
        const float* cbp0 = cb + (size_t)n0 * DIM + 2 * h;
        const float* cbp1 = cb + (size_t)n1 * DIM + 2 * h;

        v8f acc0 = {};
        v8f acc1 = {};
#pragma unroll
        for (int kk = 0; kk < 64; ++kk) {
            v2f b0 = *(const v2f*)(cbp0 + 4 * kk);
            v2f b1 = *(const v2f*)(cbp1 + 4 * kk);
            // D = A x B + C  (16x16x4, f32)
            acc0 = __builtin_amdgcn_wmma_f32_16x16x4_f32(
                false, a[kk], false, b0, (short)0, acc0, false, false);
            acc1 = __builtin_amdgcn_wmma_f32_16x16x4_f32(
                false, a[kk], false, b1, (short)0, acc1, false, false);
        }

        // score = c2[n] - 2*dot   (x^2 term is constant per row -> irrelevant to argmin)
        const float cc0 = c2[n0];
        const float cc1 = c2[n1];
#pragma unroll
        for (int r = 0; r < 8; ++r) {
            float s0 = cc0 - 2.0f * acc0[r];
            float s1 = cc1 - 2.0f * acc1[r];
            if (s0 < minv[r]) { minv[r] = s0; mini[r] = n0; }
            if (s1 < minv[r]) { minv[r] = s1; mini[r] = n1; }
        }
    }

    // Cross-lane argmin within each 16-lane half (C/D layout: lanes 0-15 hold
    // rows 0-7, lanes 16-31 hold rows 8-15; column = lane%16).
#pragma unroll
    for (int r = 0; r < 8; ++r) {
        float v = minv[r];
        int   i = mini[r];
#pragma unroll
        for (int m = 8; m >= 1; m >>= 1) {
            float ov = __shfl_xor(v, m, 16);
            int   oi = __shfl_xor(i, m, 16);
            if (ov < v || (ov == v && oi < i)) { v = ov; i = oi; }
        }
        minv[r] = v;
        mini[r] = i;
    }

    if (lr == 0) {
        const int rowOff = rowBase + h * 8;   // lane 0 -> rows 0-7, lane 16 -> rows 8-15
#pragma unroll
        for (int r = 0; r < 8; ++r)
            out_idx_f[rowOff + r] = (float)mini[r];
    }
}

// ---------------- kernel 3: gather quantized + per-block squared-error ----------------
// Block = 256 threads handles 4 rows (64 threads/row, one float4 per thread).
__global__ __launch_bounds__(256) void vq_gather_kernel(const float* __restrict__ x,
                                                        const float* __restrict__ cb,
                                                        const float* __restrict__ idx_f,
                                                        float* __restrict__ out_q,
                                                        float* __restrict__ partial) {
    __shared__ float sred[8];
    const int t    = threadIdx.x;
    const int rsub = t >> 6;                       // 0..3
    const int c4   = t & 63;                       // float4 slot within row
    const int row  = blockIdx.x * 4 + rsub;

    const int idx = (int)idx_f[row];
    const float4 q4 = *(const float4*)(cb + (size_t)idx * DIM + c4 * 4);
    const float4 x4 = *(const float4*)(x + (size_t)row * DIM + c4 * 4);
    *(float4*)(out_q + (size_t)row * DIM + c4 * 4) = q4;

    float dx = q4.x - x4.x, dy = q4.y - x4.y, dz = q4.z - x4.z, dw = q4.w - x4.w;
    float s = dx * dx + dy * dy + dz * dz + dw * dw;
#pragma unroll
    for (int m = 16; m >= 1; m >>= 1) s += __shfl_xor(s, m, 32);
    if ((t & 31) == 0) sred[t >> 5] = s;
    __syncthreads();
    if (t == 0) {
        float b = 0.0f;
#pragma unroll
        for (int w = 0; w < 8; ++w) b += sred[w];
        partial[blockIdx.x] = b;
    }
}

// ---------------- kernel 4: deterministic final loss reduction ----------------
__global__ __launch_bounds__(256) void vq_loss_kernel(const float* __restrict__ partial,
                                                      float* __restrict__ out_loss) {
    __shared__ float sred[256];
    float s = 0.0f;
    for (int i = threadIdx.x; i < (NROWS / 4); i += 256) s += partial[i];
    sred[threadIdx.x] = s;
    __syncthreads();
    for (int step = 128; step >= 1; step >>= 1) {
        if (threadIdx.x < step) sred[threadIdx.x] += sred[threadIdx.x + step];
        __syncthreads();
    }
    if (threadIdx.x == 0) {
        // vq_loss + 0.25 * commitment_loss; both equal mean((q-x)^2) in value
        out_loss[0] = 1.25f * sred[0] / ((float)NROWS * (float)DIM);
    }
}

extern "C" void kernel_launch(void* const* d_in, const int* in_sizes, int n_in,
                              void* d_out, int out_size, void* d_ws, size_t ws_size,
                              hipStream_t stream) {
    const float* x  = (const float*)d_in[0];   // [65536, 256]
    const float* cb = (const float*)d_in[1];   // [1024, 256]

    float* out      = (float*)d_out;
    float* out_q    = out;                      // 65536*256
    float* out_loss = out + (size_t)NROWS * DIM;         // 1
    float* out_idx  = out + (size_t)NROWS * DIM + 1;     // 65536 (as float values)

    float* c2      = (float*)d_ws;              // 1024 floats
    float* partial = c2 + KCODES;               // 16384 floats

    vq_c2_kernel    <<<KCODES / 256, 256, 0, stream>>>(cb, c2);
    vq_argmin_kernel<<<NROWS / 128, 256, 0, stream>>>(x, cb, c2, out_idx);
    vq_gather_kernel<<<NROWS / 4, 256, 0, stream>>>(x, cb, out_idx, out_q, partial);
    vq_loss_kernel  <<<1, 256, 0, stream>>>(partial, out_loss);
}